// MultiViewConsistencyLoss_13864154431899
// MI455X (gfx1250) — compile-verified
//
#include <hip/hip_runtime.h>

// Problem constants from the reference: V=4, B=32768, J=20.
static constexpr int NV = 4;
static constexpr int NB = 32768;
static constexpr int NJ = 20;
static constexpr int NPROB = (NV - 1) * NB;   // 98304 independent Procrustes problems
static constexpr int BLOCK = 256;             // 8 waves (wave32) per workgroup
static constexpr int NBLK = NPROB / BLOCK;    // 384 (exact)

typedef __attribute__((ext_vector_type(2))) float v2f;
typedef __attribute__((ext_vector_type(8))) float v8f;

// ---------------------------------------------------------------------------
// Kernel 1: one thread = one (view v in 1..3, batch b) Procrustes problem.
// Closed form: loss = (var2 - T^2/var1) / (3n), T = s1+s2+sign(det K)*s3,
// where s_i are singular values of K = X1^T X2 (3x3), obtained as sqrt of the
// eigenvalues of M = K^T K via the trigonometric symmetric-3x3 eigen solver.
// f64 internal math (cost is negligible; kernel is HBM-bound at ~42MB total).
// ---------------------------------------------------------------------------
__global__ __launch_bounds__(BLOCK)
void procrustes_kernel(const float* __restrict__ joints,   // (V,B,J,3) f32
                       const int* __restrict__ tw,         // (V,B,J,1) i32
                       float* __restrict__ partials)       // (NBLK,) f32
{
    const int tid = threadIdx.x;
    const int p   = blockIdx.x * BLOCK + tid;          // 0..NPROB-1, exact grid
    const int v   = p / NB + 1;                        // 1..3
    const int b   = p - (v - 1) * NB;

    // 16B-aligned vector loads: 60 floats = 15 x float4, 20 ints = 5 x int4.
    const float4* S1 = (const float4*)(joints + ((size_t)v * NB + b) * (NJ * 3));
    const float4* S2 = (const float4*)(joints + (size_t)b * (NJ * 3));
    const int4*   WV = (const int4*)(tw + ((size_t)v * NB + b) * NJ);
    const int4*   W0 = (const int4*)(tw + (size_t)b * NJ);

    float a1[NJ * 3], a2[NJ * 3];
#pragma unroll
    for (int i = 0; i < 15; ++i) {
        float4 t = S1[i];
        a1[4 * i + 0] = t.x; a1[4 * i + 1] = t.y;
        a1[4 * i + 2] = t.z; a1[4 * i + 3] = t.w;
    }
#pragma unroll
    for (int i = 0; i < 15; ++i) {
        float4 t = S2[i];
        a2[4 * i + 0] = t.x; a2[4 * i + 1] = t.y;
        a2[4 * i + 2] = t.z; a2[4 * i + 3] = t.w;
    }

    unsigned maskbits = 0;
#pragma unroll
    for (int i = 0; i < 5; ++i) {
        int4 wa = WV[i];
        int4 wb = W0[i];
        if (wa.x * wb.x > 0) maskbits |= 1u << (4 * i + 0);
        if (wa.y * wb.y > 0) maskbits |= 1u << (4 * i + 1);
        if (wa.z * wb.z > 0) maskbits |= 1u << (4 * i + 2);
        if (wa.w * wb.w > 0) maskbits |= 1u << (4 * i + 3);
    }

    // Pass 1: n, masked means.
    double n = 0.0;
    double mu1x = 0, mu1y = 0, mu1z = 0, mu2x = 0, mu2y = 0, mu2z = 0;
#pragma unroll
    for (int j = 0; j < NJ; ++j) {
        double m = ((maskbits >> j) & 1u) ? 1.0 : 0.0;
        n += m;
        mu1x += m * a1[3 * j + 0]; mu1y += m * a1[3 * j + 1]; mu1z += m * a1[3 * j + 2];
        mu2x += m * a2[3 * j + 0]; mu2y += m * a2[3 * j + 1]; mu2z += m * a2[3 * j + 2];
    }
    const double inv_n = 1.0 / n;                      // n==0 -> NaN, matches ref
    mu1x *= inv_n; mu1y *= inv_n; mu1z *= inv_n;
    mu2x *= inv_n; mu2y *= inv_n; mu2z *= inv_n;

    // Pass 2: var1, var2, K = X1^T X2.
    double var1 = 0, var2 = 0;
    double K00 = 0, K01 = 0, K02 = 0, K10 = 0, K11 = 0, K12 = 0, K20 = 0, K21 = 0, K22 = 0;
#pragma unroll
    for (int j = 0; j < NJ; ++j) {
        double m = ((maskbits >> j) & 1u) ? 1.0 : 0.0;
        double x0 = m * ((double)a1[3 * j + 0] - mu1x);
        double x1 = m * ((double)a1[3 * j + 1] - mu1y);
        double x2 = m * ((double)a1[3 * j + 2] - mu1z);
        double y0 = m * ((double)a2[3 * j + 0] - mu2x);
        double y1 = m * ((double)a2[3 * j + 1] - mu2y);
        double y2 = m * ((double)a2[3 * j + 2] - mu2z);
        var1 += x0 * x0 + x1 * x1 + x2 * x2;
        var2 += y0 * y0 + y1 * y1 + y2 * y2;
        K00 += x0 * y0; K01 += x0 * y1; K02 += x0 * y2;
        K10 += x1 * y0; K11 += x1 * y1; K12 += x1 * y2;
        K20 += x2 * y0; K21 += x2 * y1; K22 += x2 * y2;
    }

    const double detK = K00 * (K11 * K22 - K12 * K21)
                      - K01 * (K10 * K22 - K12 * K20)
                      + K02 * (K10 * K21 - K11 * K20);

    // M = K^T K (symmetric PSD).
    const double M00 = K00 * K00 + K10 * K10 + K20 * K20;
    const double M01 = K00 * K01 + K10 * K11 + K20 * K21;
    const double M02 = K00 * K02 + K10 * K12 + K20 * K22;
    const double M11 = K01 * K01 + K11 * K11 + K21 * K21;
    const double M12 = K01 * K02 + K11 * K12 + K21 * K22;
    const double M22 = K02 * K02 + K12 * K12 + K22 * K22;

    // Trigonometric closed-form eigenvalues of symmetric 3x3.
    const double q  = (M00 + M11 + M22) * (1.0 / 3.0);
    const double p1 = M01 * M01 + M02 * M02 + M12 * M12;
    const double aq = M00 - q, bq = M11 - q, cq = M22 - q;
    const double p2 = aq * aq + bq * bq + cq * cq + 2.0 * p1;
    double l1, l2, l3;
    if (p2 <= 0.0) {
        l1 = l2 = l3 = q;                              // M == q*I
    } else {
        const double pp  = sqrt(p2 * (1.0 / 6.0));
        const double ip  = 1.0 / pp;
        const double b00 = aq * ip, b11 = bq * ip, b22 = cq * ip;
        const double b01 = M01 * ip, b02 = M02 * ip, b12 = M12 * ip;
        double r = 0.5 * (b00 * (b11 * b22 - b12 * b12)
                        - b01 * (b01 * b22 - b12 * b02)
                        + b02 * (b01 * b12 - b11 * b02));
        r = fmin(1.0, fmax(-1.0, r));
        const double phi = acos(r) * (1.0 / 3.0);
        l1 = q + 2.0 * pp * cos(phi);                       // largest
        l3 = q + 2.0 * pp * cos(phi + 2.0943951023931953);  // smallest (+2pi/3)
        l2 = 3.0 * q - l1 - l3;
    }
    const double s1 = sqrt(fmax(l1, 0.0));
    const double s2 = sqrt(fmax(l2, 0.0));
    const double s3 = sqrt(fmax(l3, 0.0));
    const double sgn = (detK > 0.0) ? 1.0 : ((detK < 0.0) ? -1.0 : 0.0);
    const double T = s1 + s2 + sgn * s3;

    const double loss = (var2 - (T * T) / var1) / (3.0 * n);

    // Deterministic in-block tree reduction.
    __shared__ float sdata[BLOCK];
    sdata[tid] = (float)loss;
    __syncthreads();
#pragma unroll
    for (int s = BLOCK / 2; s > 0; s >>= 1) {
        if (tid < s) sdata[tid] += sdata[tid + s];
        __syncthreads();
    }
    if (tid == 0) partials[blockIdx.x] = sdata[0];
}

// ---------------------------------------------------------------------------
// Kernel 2: single wave32. Each lane accumulates a strided (deterministic)
// partial sum, then V_WMMA_F32_16X16X4_F32 with A=ones reduces the 64 B-slots:
// every row of D holds the column sums of B, so summing one 16-lane row of D
// yields the grand total regardless of the exact B VGPR slot mapping.
// ---------------------------------------------------------------------------
__global__ __launch_bounds__(32)
void reduce_wmma_kernel(const float* __restrict__ partials, float* __restrict__ out)
{
    const int lane = threadIdx.x;                  // 32 lanes, all active
    float acc = 0.0f;
#pragma unroll
    for (int i = 0; i < NBLK / 32; ++i)            // uniform trip count: no divergence
        acc += partials[i * 32 + lane];

    v2f amat; amat.x = 1.0f; amat.y = 1.0f;        // A = ones(16x4)
    v2f bmat; bmat.x = acc;  bmat.y = 0.0f;        // 32 partials + 32 zeros fill B(4x16)
    v8f c = {};
    // 8 args: (neg_a, A, neg_b, B, c_mod, C, reuse_a, reuse_b)
    v8f d = __builtin_amdgcn_wmma_f32_16x16x4_f32(
        false, amat, false, bmat, (short)0, c, false, false);

    // d[0] on lanes 0..15 = row 0 of D = column sums; xor-reduce within 16 lanes.
    float r = d[0];
    r += __shfl_xor(r, 1, 32);
    r += __shfl_xor(r, 2, 32);
    r += __shfl_xor(r, 4, 32);
    r += __shfl_xor(r, 8, 32);

    if (lane == 0) out[0] = r * (1.0f / (float)NPROB);
}

extern "C" void kernel_launch(void* const* d_in, const int* in_sizes, int n_in,
                              void* d_out, int out_size, void* d_ws, size_t ws_size,
                              hipStream_t stream)
{
    const float* joints = (const float*)d_in[0];   // (4,32768,20,3) f32
    const int*   tw     = (const int*)d_in[1];     // (4,32768,20,1) i32
    float* partials = (float*)d_ws;                // NBLK floats of scratch

    procrustes_kernel<<<NBLK, BLOCK, 0, stream>>>(joints, tw, partials);
    reduce_wmma_kernel<<<1, 32, 0, stream>>>(partials, (float*)d_out);
}